// Model_36163624632743
// MI455X (gfx1250) — compile-verified
//
#include <hip/hip_runtime.h>
#include <hip/hip_bf16.h>
#include <math.h>

typedef __attribute__((ext_vector_type(2))) float v2f;
typedef __attribute__((ext_vector_type(8))) float v8f;

#define HN 2048

// ---------------------------------------------------------------------------
// Kernel 1: feature MLP (462->66->66->H) + embedding (34->H) -> hid (H floats)
// Tiny (~0.4 MFLOP); one workgroup.
// ---------------------------------------------------------------------------
__global__ __launch_bounds__(256)
void prep_kernel(const float* __restrict__ h0,
                 const float* __restrict__ feature,
                 const float* __restrict__ emb_w, const float* __restrict__ emb_b,
                 const float* __restrict__ f1_w, const float* __restrict__ f1_b,
                 const float* __restrict__ f2_w, const float* __restrict__ f2_b,
                 const float* __restrict__ f3_w, const float* __restrict__ f3_b,
                 float* __restrict__ h_out)
{
    __shared__ float s_feat[462];
    __shared__ float s_fd1[66];
    __shared__ float s_fd2[66];
    __shared__ float s_h0[34];

    const int tid = threadIdx.x;
    for (int i = tid; i < 462; i += 256) s_feat[i] = feature[i];
    if (tid < 34) s_h0[tid] = h0[tid];
    __syncthreads();

    if (tid < 66) {
        float acc = f1_b[tid];
        const float* w = f1_w + tid * 462;
        for (int k = 0; k < 462; ++k) acc += s_feat[k] * w[k];
        s_fd1[tid] = acc;
    }
    __syncthreads();

    if (tid < 66) {
        float acc = f2_b[tid];
        const float* w = f2_w + tid * 66;
        for (int k = 0; k < 66; ++k) acc += s_fd1[k] * w[k];
        s_fd2[tid] = acc;
    }
    __syncthreads();

    for (int j = tid; j < HN; j += 256) {
        float acc = emb_b[j] + f3_b[j];
        const float* we = emb_w + j * 34;
        const float* wf = f3_w + j * 66;
        for (int k = 0; k < 34; ++k) acc += s_h0[k] * we[k];
        for (int k = 0; k < 66; ++k) acc += s_fd2[k] * wf[k];
        h_out[j] = acc;
    }
}

// ---------------------------------------------------------------------------
// Kernel 2 (x7): one GRU step. Bandwidth-bound matvec gh = W_hh[s] * h via
// V_WMMA_F32_16X16X4_F32: h broadcast in A (from LDS), weight tile in B
// (streamed: one global_load_b64 per lane per wmma = 256B/wave per wmma).
// Block: 12 waves = 3 gates x 4 K-slices; 128 blocks cover j = 0..2047.
// Dual accumulator chains (c0/c1) keep 8 global loads in flight per wave.
// ---------------------------------------------------------------------------
__global__ __launch_bounds__(384)
void gru_step_kernel(const float* __restrict__ xt,
                     const float* __restrict__ w_ih,
                     const float* __restrict__ w_hh,
                     const float* __restrict__ b_ih,
                     const float* __restrict__ b_hh,
                     const float* __restrict__ h_in,
                     float* __restrict__ h_out,
                     int step)
{
    __shared__ __align__(16) float lds_h[HN];
    __shared__ float partials[3][4][16];

    const int tid = threadIdx.x;
    // Stage h into LDS (coalesced).
    for (int i = tid; i < HN; i += 384) lds_h[i] = h_in[i];
    __syncthreads();

    const int wave   = tid >> 5;
    const int lane   = tid & 31;
    const int gate   = wave % 3;       // 0=r, 1=z, 2=n
    const int kslice = wave / 3;       // 0..3, each covers 512 of K
    const int n_idx  = lane & 15;      // output column within the 16-wide tile
    const int khalf  = lane >> 4;      // lanes 16-31 hold K+2,K+3

    const size_t w_off = (size_t)step * 3 * HN * HN;
    // B-tile row for this lane: W_hh[gate*H + tile*16 + n][:]
    const float* wrow = w_hh + w_off +
                        (size_t)(gate * HN + blockIdx.x * 16 + n_idx) * HN;

    const int klo = kslice * (HN / 4);         // 512-wide K slice
    const float* gp = wrow + klo + 2 * khalf;  // per-lane global base
    const float* lp = &lds_h[klo + 2 * khalf]; // per-lane LDS base

    v8f c0 = {};
    v8f c1 = {};
    #pragma unroll 4
    for (int k = 0; k < HN / 4; k += 8) {
        v2f b0 = *(const v2f*)(gp + k);
        v2f a0 = *(const v2f*)(lp + k);
        c0 = __builtin_amdgcn_wmma_f32_16x16x4_f32(
                false, a0, false, b0, (short)0, c0, false, false);
        v2f b1 = *(const v2f*)(gp + k + 4);
        v2f a1 = *(const v2f*)(lp + k + 4);
        c1 = __builtin_amdgcn_wmma_f32_16x16x4_f32(
                false, a1, false, b1, (short)0, c1, false, false);
    }
    // All D rows equal (A rows identical): lane n (<16) holds output n in c[0].
    if (lane < 16) partials[gate][kslice][n_idx] = c0[0] + c1[0];
    __syncthreads();

    if (tid < 16) {
        const int jg = blockIdx.x * 16 + tid;
        const float x = xt[step];
        const float* bih = b_ih + step * 3 * HN;
        const float* bhh = b_hh + step * 3 * HN;
        const float* wih = w_ih + step * 3 * HN;   // shape (3H,1) flattened

        float hr = bhh[jg];
        float hz = bhh[HN + jg];
        float hn = bhh[2 * HN + jg];
        #pragma unroll
        for (int ks = 0; ks < 4; ++ks) {
            hr += partials[0][ks][tid];
            hz += partials[1][ks][tid];
            hn += partials[2][ks][tid];
        }
        float ir  = x * wih[jg]           + bih[jg];
        float iz  = x * wih[HN + jg]      + bih[HN + jg];
        float inn = x * wih[2 * HN + jg]  + bih[2 * HN + jg];

        float r = 1.0f / (1.0f + __expf(-(ir + hr)));
        float z = 1.0f / (1.0f + __expf(-(iz + hz)));
        float n = tanhf(inn + r * hn);
        h_out[jg] = (1.0f - z) * n + z * lds_h[jg];
    }
}

// ---------------------------------------------------------------------------
// Kernel 3: out = [h7, restrict, number] . out_w + out_b
// ---------------------------------------------------------------------------
__global__ __launch_bounds__(256)
void out_kernel(const float* __restrict__ h7,
                const float* __restrict__ number,
                const float* __restrict__ restr,
                const float* __restrict__ out_w,
                const float* __restrict__ out_b,
                float* __restrict__ out)
{
    __shared__ float red[256];
    const int tid = threadIdx.x;
    float acc = 0.0f;
    for (int j = tid; j < HN; j += 256) acc += h7[j] * out_w[j];
    red[tid] = acc;
    __syncthreads();
    for (int s = 128; s > 0; s >>= 1) {
        if (tid < s) red[tid] += red[tid + s];
        __syncthreads();
    }
    if (tid == 0)
        out[0] = red[0] + restr[0] * out_w[HN] + number[0] * out_w[HN + 1] + out_b[0];
}

// ---------------------------------------------------------------------------
extern "C" void kernel_launch(void* const* d_in, const int* in_sizes, int n_in,
                              void* d_out, int out_size, void* d_ws, size_t ws_size,
                              hipStream_t stream) {
    const float* xt      = (const float*)d_in[0];
    const float* h0      = (const float*)d_in[1];
    const float* feature = (const float*)d_in[2];
    const float* number  = (const float*)d_in[3];
    const float* restr   = (const float*)d_in[4];
    const float* w_ih    = (const float*)d_in[5];
    const float* w_hh    = (const float*)d_in[6];
    const float* b_ih    = (const float*)d_in[7];
    const float* b_hh    = (const float*)d_in[8];
    const float* emb_w   = (const float*)d_in[9];
    const float* emb_b   = (const float*)d_in[10];
    const float* f1_w    = (const float*)d_in[11];
    const float* f1_b    = (const float*)d_in[12];
    const float* f2_w    = (const float*)d_in[13];
    const float* f2_b    = (const float*)d_in[14];
    const float* f3_w    = (const float*)d_in[15];
    const float* f3_b    = (const float*)d_in[16];
    const float* out_w   = (const float*)d_in[17];
    const float* out_b   = (const float*)d_in[18];

    float* hbuf0 = (float*)d_ws;        // double-buffered hidden state
    float* hbuf1 = hbuf0 + HN;

    prep_kernel<<<1, 256, 0, stream>>>(h0, feature, emb_w, emb_b,
                                       f1_w, f1_b, f2_w, f2_b, f3_w, f3_b,
                                       hbuf0);

    for (int s = 0; s < 7; ++s) {
        float* hin  = (s & 1) ? hbuf1 : hbuf0;
        float* hout = (s & 1) ? hbuf0 : hbuf1;
        gru_step_kernel<<<128, 384, 0, stream>>>(xt, w_ih, w_hh, b_ih, b_hh,
                                                 hin, hout, s);
    }

    // after 7 steps the state sits in hbuf1
    out_kernel<<<1, 256, 0, stream>>>(hbuf1, number, restr, out_w, out_b,
                                      (float*)d_out);
}